// VectorQuantizer_44667659878737
// MI455X (gfx1250) — compile-verified
//
#include <hip/hip_runtime.h>

typedef __attribute__((ext_vector_type(2))) float v2f;
typedef __attribute__((ext_vector_type(8))) float v8f;

#define NUM_EMB 8192
#define DIM 64
#define NROWS 32768
#define WAVES_PER_BLOCK 8
#define ROWS_PER_BLOCK 128          // 8 waves * 16 rows
#define BLK_CODES 64                // codes staged per LDS block (4 WMMA tiles)
#define LDS_ROW 68                  // 64 + 4 pad: bijective banks for b64 reads, 16B rows
#define BUF_FLOATS (BLK_CODES * LDS_ROW)
#define NBLK (NUM_EMB / BLK_CODES)  // 128

// Kernel A: wn[k] = ||w_k||^2
__global__ __launch_bounds__(256) void vq_wnorm_kernel(const float* __restrict__ w,
                                                       float* __restrict__ wn) {
    int k = blockIdx.x * blockDim.x + threadIdx.x;
    if (k >= NUM_EMB) return;
    const float4* p = (const float4*)(w + k * DIM);
    float s = 0.0f;
#pragma unroll
    for (int i = 0; i < DIM / 4; ++i) {
        float4 v = p[i];
        s = fmaf(v.x, v.x, s);
        s = fmaf(v.y, v.y, s);
        s = fmaf(v.z, v.z, s);
        s = fmaf(v.w, v.w, s);
    }
    wn[k] = s;
}

// Async global->LDS copy, one 16B chunk (CDNA5 path, ASYNCcnt-tracked)
__device__ __forceinline__ void async_b128_to_lds(unsigned lds_byte_off,
                                                  const float* gaddr) {
    asm volatile("global_load_async_to_lds_b128 %0, %1, off"
                 :: "v"(lds_byte_off), "v"(gaddr)
                 : "memory");
}

// Stage a 64-code x 64-dim W block into LDS (padded rows), 4 chunks per thread
__device__ __forceinline__ void stage_block_async(const float* __restrict__ w,
                                                  int k0, unsigned lds_base_bytes,
                                                  int tid) {
#pragma unroll
    for (int j = 0; j < 4; ++j) {
        int c = tid + j * 256;         // 1024 float4-chunks total
        int row = c >> 4;              // 0..63
        int q   = c & 15;              // 0..15
        unsigned loff = lds_base_bytes + (unsigned)((row * LDS_ROW + q * 4) * 4);
        async_b128_to_lds(loff, w + (size_t)(k0 + row) * DIM + q * 4);
    }
}

// Kernel B: WMMA distance scores + argmin + gather + loss partials.
__global__ __launch_bounds__(256) void vq_argmin_kernel(
    const float* __restrict__ x, const float* __restrict__ w,
    const float* __restrict__ wn, float* __restrict__ outq,
    float* __restrict__ outidx, float* __restrict__ lpart) {
    __shared__ float wtile[2 * BUF_FLOATS];   // double-buffered W block
    __shared__ float lred[WAVES_PER_BLOCK];

    const int tid  = threadIdx.x;
    const int lane = tid & 31;
    const int wave = tid >> 5;
    const int m  = lane & 15;   // A-row (code) / B-col (input row) within tile
    const int hi = lane >> 4;   // selects K pair {2hi, 2hi+1} per fragment

    const int row0 = blockIdx.x * ROWS_PER_BLOCK + wave * 16;

    const unsigned lds0 = (unsigned)(size_t)(void*)&wtile[0];

    // B fragments held in registers for the whole K loop:
    // lane holds X[row0+m][kk*4 + 2hi + {0,1}]
    v2f xb[16];
    const float* xrow = x + (size_t)(row0 + m) * DIM + 2 * hi;
#pragma unroll
    for (int kk = 0; kk < 16; ++kk)
        xb[kk] = *(const v2f*)(xrow + kk * 4);

    float best = __builtin_inff();
    int   bidx = 0;

    // Prologue: async-stage block 0 into buffer 0
    stage_block_async(w, 0, lds0, tid);

#pragma unroll 1
    for (int blk = 0; blk < NBLK; ++blk) {
        const int cur = blk & 1;
        __syncthreads();   // all waves done reading buf[cur^1] -> safe to overwrite
        if (blk + 1 < NBLK) {
            stage_block_async(w, (blk + 1) * BLK_CODES,
                              lds0 + (unsigned)((cur ^ 1) * (BUF_FLOATS * 4)), tid);
            // async loads complete in order: <=4 outstanding => block blk resident
            asm volatile("s_wait_asynccnt 0x4" ::: "memory");
        } else {
            asm volatile("s_wait_asynccnt 0x0" ::: "memory");
        }
        __syncthreads();   // block blk visible to all waves

        const float* buf = &wtile[cur * BUF_FLOATS];
        const int k0 = blk * BLK_CODES;
#pragma unroll
        for (int t = 0; t < 4; ++t) {
            // Load all 16 A fragments first (conflict-free b64), then WMMA chain
            v2f afr[16];
#pragma unroll
            for (int kk = 0; kk < 16; ++kk)
                afr[kk] = *(const v2f*)(buf + (16 * t + m) * LDS_ROW + kk * 4 + 2 * hi);
            v8f acc = {};
#pragma unroll
            for (int kk = 0; kk < 16; ++kk)
                acc = __builtin_amdgcn_wmma_f32_16x16x4_f32(
                    false, afr[kk], false, xb[kk], (short)0, acc, false, false);

            // score = ||w||^2 - 2*dot ; in-lane argmin, ascending code index
            const float* wnp = wn + k0 + 16 * t + 8 * hi;
#pragma unroll
            for (int r = 0; r < 8; ++r) {
                float s = fmaf(-2.0f, acc[r], wnp[r]);
                if (s < best) { best = s; bidx = k0 + 16 * t + 8 * hi + r; }
            }
        }
    }

    // Merge the two lane halves (codes split 0..7 / 8..15 per tile)
    {
        float ov = __shfl_xor(best, 16, 32);
        int   oi = __shfl_xor(bidx, 16, 32);
        if (ov < best || (ov == best && oi < bidx)) { best = ov; bidx = oi; }
    }

    // Indices (as float, matching tuple-concat promotion)
    if (hi == 0) outidx[row0 + m] = (float)bidx;

    // Gather quantized rows, write output, accumulate commitment-loss partial
    float lacc = 0.0f;
#pragma unroll 1
    for (int r = 0; r < 16; ++r) {
        int idx = __shfl(bidx, r, 32);
        v2f q  = *(const v2f*)(w + (size_t)idx * DIM + lane * 2);
        v2f xi = *(const v2f*)(x + (size_t)(row0 + r) * DIM + lane * 2);
        *(v2f*)(outq + (size_t)(row0 + r) * DIM + lane * 2) = q;
        float dx = q.x - xi.x, dy = q.y - xi.y;
        lacc = fmaf(dx, dx, lacc);
        lacc = fmaf(dy, dy, lacc);
    }
#pragma unroll
    for (int off = 16; off > 0; off >>= 1)
        lacc += __shfl_xor(lacc, off, 32);
    if (lane == 0) lred[wave] = lacc;
    __syncthreads();
    if (tid == 0) {
        float s = 0.0f;
#pragma unroll
        for (int i = 0; i < WAVES_PER_BLOCK; ++i) s += lred[i];
        lpart[blockIdx.x] = s;
    }
}

// Kernel C: reduce 256 block partials -> scalar loss (deterministic, no atomics)
__global__ __launch_bounds__(256) void vq_loss_kernel(const float* __restrict__ lpart,
                                                      float* __restrict__ out_loss) {
    __shared__ float s[256];
    int t = threadIdx.x;
    s[t] = lpart[t];
    __syncthreads();
    for (int off = 128; off > 0; off >>= 1) {
        if (t < off) s[t] += s[t + off];
        __syncthreads();
    }
    if (t == 0)
        out_loss[0] = 0.25f * s[0] / (float)(NROWS * DIM);
}

extern "C" void kernel_launch(void* const* d_in, const int* in_sizes, int n_in,
                              void* d_out, int out_size, void* d_ws, size_t ws_size,
                              hipStream_t stream) {
    const float* x = (const float*)d_in[0];   // [8,4096,64] f32
    const float* w = (const float*)d_in[1];   // [8192,64]   f32
    float* out  = (float*)d_out;
    float* outq   = out;                       // 2,097,152 floats
    float* outl   = out + (size_t)NROWS * DIM; // 1 float
    float* outidx = outl + 1;                  // 32,768 floats

    float* wn    = (float*)d_ws;               // 8192 floats
    float* lpart = wn + NUM_EMB;               // 256 floats

    vq_wnorm_kernel<<<NUM_EMB / 256, 256, 0, stream>>>(w, wn);
    vq_argmin_kernel<<<NROWS / ROWS_PER_BLOCK, 256, 0, stream>>>(
        x, w, wn, outq, outidx, lpart);
    vq_loss_kernel<<<1, 256, 0, stream>>>(lpart, outl);
}